// Matching_Reducer_36464272343583
// MI455X (gfx1250) — compile-verified
//
#include <hip/hip_runtime.h>
#include <hip/hip_bf16.h>
#include <math.h>

// Problem constants (match reference setup_inputs)
#define BB 16
#define HH 50
#define SS 256
#define DD 256
#define KK 32

typedef float v2f __attribute__((ext_vector_type(2)));
typedef float v8f __attribute__((ext_vector_type(8)));

// ---------------------------------------------------------------------------
// Kernel 0: q_n[b] = user_repr[b,0,:] / max(||.||, 1e-12)
// ---------------------------------------------------------------------------
__global__ __launch_bounds__(256) void qn_kernel(const float* __restrict__ user,
                                                 float* __restrict__ qn) {
    const int b = blockIdx.x;
    const int d = threadIdx.x;
    __shared__ float red[256];
    float v = user[b * DD + d];
    red[d] = v * v;
    __syncthreads();
#pragma unroll
    for (int s = 128; s > 0; s >>= 1) {
        if (d < s) red[d] += red[d + s];
        __syncthreads();
    }
    float n = fmaxf(sqrtf(red[0]), 1e-12f);
    qn[b * DD + d] = v / n;
}

// ---------------------------------------------------------------------------
// Kernel 1: scores[bh][s] = dot(sel[bh][1+s], q_n[b]) / max(||sel row||,1e-12)
//           (masked to -inf when invalid; s==255 pad entry is -inf)
// One wave (32 lanes) per 16-row tile. Dot products via V_WMMA_F32_16X16X4_F32:
//   A 16x4: lane m (m<16) holds row m k={0,1}; lane m+16 holds row m k={2,3}
//   B 4x16: only column N==0 nonzero -> lane 0 carries q[k0],q[k1],
//                                       lane 16 carries q[k2],q[k3]
//   D 16x16: column N==0 -> lane 0 holds M=0..7 in c[0..7], lane 16 M=8..15.
// Row ||.||^2 comes free from the same A loads (lane + lane^16 partials).
// B operand built branch-free: every lane does the same ds_load_b64 of the
// q chunk, then multiplies by a loop-invariant 0/1 lane mask (no EXEC
// save/restore in the hot loop).
// ---------------------------------------------------------------------------
__global__ __launch_bounds__(32) void scores_kernel(const float* __restrict__ sel,
                                                    const float* __restrict__ qn,
                                                    const int* __restrict__ refined,
                                                    float* __restrict__ scores) {
    const int bh   = blockIdx.x;         // b*H + h
    const int tile = blockIdx.y;         // 0..15 (16 rows each)
    const int lane = threadIdx.x;        // wave32
    const int b    = bh / HH;

    __shared__ float qs[DD];
    __shared__ float dots[16];
    __shared__ float nsh[32];

    // stage normalized query in LDS
    for (int i = lane; i < DD; i += 32) qs[i] = qn[b * DD + i];
    __syncthreads();

    const int rowInTile = lane & 15;
    const int khalf     = lane >> 4;     // 0: k={0,1}, 1: k={2,3}
    int s = tile * 16 + rowInTile;       // slice index 0..255 (255 = pad)
    int srow = (s < 255) ? s : 254;      // clamp so EXEC stays all-ones for WMMA
    const float* rowp = sel + ((size_t)bh * SS + 1 + srow) * DD + khalf * 2;
    const float* qrow = qs + khalf * 2;

    v8f c = {};                          // 16x16 f32 accumulator
    float nacc = 0.0f;
    // lanes 0 and 16 feed B column N==0; all others contribute zeros
    const float qmask = (rowInTile == 0) ? 1.0f : 0.0f;

#pragma unroll 8
    for (int kc = 0; kc < DD / 4; ++kc) {
        float2 av = *(const float2*)(rowp + kc * 4);   // global_load_b64
        float2 qv = *(const float2*)(qrow + kc * 4);   // ds_load_b64 (uniform)
        v2f a;  a.x  = av.x;         a.y  = av.y;
        v2f bv; bv.x = qv.x * qmask; bv.y = qv.y * qmask;
        nacc += av.x * av.x + av.y * av.y;
        c = __builtin_amdgcn_wmma_f32_16x16x4_f32(
                /*neg_a=*/false, a, /*neg_b=*/false, bv,
                /*c_mod=*/(short)0, c, /*reuse_a=*/false, /*reuse_b=*/false);
    }

    nsh[lane] = nacc;
    if (lane == 0) {
#pragma unroll
        for (int j = 0; j < 8; ++j) dots[j] = c[j];
    }
    if (lane == 16) {
#pragma unroll
        for (int j = 0; j < 8; ++j) dots[8 + j] = c[j];
    }
    __syncthreads();

    if (lane < 16) {
        float out;
        if (s >= 255) {
            out = -INFINITY;                       // pad entry
        } else {
            float n2  = nsh[lane] + nsh[lane + 16];
            float nrm = fmaxf(sqrtf(n2), 1e-12f);
            float sc  = dots[lane] / nrm;
            int   m   = refined[(size_t)bh * SS + 1 + s];
            bool valid = (m != 0) || (s < KK);
            out = valid ? sc : -INFINITY;
        }
        scores[(size_t)bh * 256 + s] = out;
    }
}

// ---------------------------------------------------------------------------
// Kernel 2: per (b,h): sorted top-32 (desc, ties -> lowest index, matching
// lax.top_k), softmax weights, mask gather, kid output. One wave per (b,h).
// ---------------------------------------------------------------------------
__global__ __launch_bounds__(32) void topk_kernel(const float* __restrict__ scores,
                                                  const float* __restrict__ am,
                                                  float* __restrict__ w_out,
                                                  int* __restrict__ kid_out_i,
                                                  float* __restrict__ out_mask,
                                                  float* __restrict__ out_kid) {
    const int bh   = blockIdx.x;
    const int lane = threadIdx.x;

    float v[8];
#pragma unroll
    for (int j = 0; j < 8; ++j) v[j] = scores[(size_t)bh * 256 + lane * 8 + j];

    __shared__ float kv[KK];
    __shared__ int   ki[KK];

    for (int it = 0; it < KK; ++it) {
        float bvv = -INFINITY;
        int   bii = 0x7fffffff;
#pragma unroll
        for (int j = 0; j < 8; ++j) {
            int idx = lane * 8 + j;
            if (v[j] > bvv || (v[j] == bvv && idx < bii)) { bvv = v[j]; bii = idx; }
        }
#pragma unroll
        for (int off = 16; off >= 1; off >>= 1) {
            float ov = __shfl_xor(bvv, off, 32);
            int   oi = __shfl_xor(bii, off, 32);
            if (ov > bvv || (ov == bvv && oi < bii)) { bvv = ov; bii = oi; }
        }
        if ((bii >> 3) == lane) v[bii & 7] = -INFINITY;   // knock out winner
        if (lane == 0) { kv[it] = bvv; ki[it] = bii; }
    }
    __syncthreads();

    // softmax over the 32 selected scores; kv[0] is the max
    float sv = kv[lane];
    int   si = ki[lane];
    float e  = expf(sv - kv[0]);
    float sum = e;
#pragma unroll
    for (int off = 16; off >= 1; off >>= 1) sum += __shfl_xor(sum, off, 32);
    float w = e / sum;

    const size_t o = (size_t)bh * KK + lane;   // == b*(H*K) + h*K + lane
    w_out[o]     = w;
    kid_out_i[o] = si;
    out_kid[o]   = (float)si;
    out_mask[o]  = am[(size_t)bh * SS + 1 + si];
}

// ---------------------------------------------------------------------------
// Kernel 3: ps_terms[b, h*K+j, :] = txt[b,h,1+kid]* w + segment_embedding[h,0,:]
// One block per (b,h,j); fully coalesced 1 KB row copy + FMA.
// ---------------------------------------------------------------------------
__global__ __launch_bounds__(256) void gather_kernel(const float* __restrict__ txt,
                                                     const float* __restrict__ seg,
                                                     const float* __restrict__ w_arr,
                                                     const int* __restrict__ kid_arr,
                                                     float* __restrict__ out) {
    const int idx = blockIdx.x;          // bh*K + j
    const int bh  = idx >> 5;
    const int h   = bh % HH;
    const int d   = threadIdx.x;
    const int kid = kid_arr[idx];
    const float w = w_arr[idx];
    const float* src = txt + ((size_t)bh * SS + 1 + kid) * DD;
    out[(size_t)idx * DD + d] = src[d] * w + seg[h * DD + d];
}

// ---------------------------------------------------------------------------
extern "C" void kernel_launch(void* const* d_in, const int* in_sizes, int n_in,
                              void* d_out, int out_size, void* d_ws, size_t ws_size,
                              hipStream_t stream) {
    const float* sel     = (const float*)d_in[0];  // news_selection_embedding (B,H,S,D)
    const float* txt     = (const float*)d_in[1];  // news_embedding          (B,H,S,D)
    const float* user    = (const float*)d_in[2];  // user_repr               (B,1,D)
    // d_in[3] = news_repr (unused by reference)
    const float* am      = (const float*)d_in[4];  // his_attn_mask           (B,H,S)
    const int*   refined = (const int*)  d_in[5];  // his_refined_mask        (B,H,S) int32
    const float* seg     = (const float*)d_in[6];  // segment_embedding       (H,1,D)

    float* out = (float*)d_out;
    float* out_ps   = out;                                   // B*H*K*D
    float* out_mask = out + (size_t)BB * HH * KK * DD;       // B*H*K
    float* out_kid  = out_mask + (size_t)BB * HH * KK;       // B*H*K (float(idx))

    // workspace layout (~1.04 MB)
    float* qn     = (float*)d_ws;                 // B*D
    float* scores = qn + BB * DD;                 // B*H*256 (padded, s=255 => -inf)
    float* warr   = scores + (size_t)BB * HH * 256;
    int*   kidw   = (int*)(warr + (size_t)BB * HH * KK);

    qn_kernel<<<BB, 256, 0, stream>>>(user, qn);
    scores_kernel<<<dim3(BB * HH, 16), 32, 0, stream>>>(sel, qn, refined, scores);
    topk_kernel<<<BB * HH, 32, 0, stream>>>(scores, am, warr, kidw, out_mask, out_kid);
    gather_kernel<<<BB * HH * KK, 256, 0, stream>>>(txt, seg, warr, kidw, out_ps);
}